// LSTMOptimizer_7550552507088
// MI455X (gfx1250) — compile-verified
//
#include <hip/hip_runtime.h>
#include <hip/hip_bf16.h>
#include <math.h>

typedef __attribute__((ext_vector_type(2))) float v2f;
typedef __attribute__((ext_vector_type(8))) float v8f;

#define HD 20      // hidden dim
#define GATES 80   // 4*HD

__device__ __forceinline__ float sigm_f(float x) { return 1.0f / (1.0f + __expf(-x)); }
__device__ __forceinline__ float tanh_f(float x) { return 2.0f / (1.0f + __expf(-2.0f * x)) - 1.0f; }

// ---------------------------------------------------------------------------
// Kernel 1: layer-0 LSTM scan (serial over N steps), one wave32.
// Gate j -> lane j&31, slot j>>5.  h[20] replicated across lanes in registers.
// Preprocess (log-mag / sign channels) + K=2 input projection fused in.
// ---------------------------------------------------------------------------
__global__ void __launch_bounds__(32)
lstm_scan0(const float* __restrict__ grad,
           const float* __restrict__ Wih0, const float* __restrict__ Whh0,
           const float* __restrict__ bih0, const float* __restrict__ bhh0,
           float* __restrict__ y0, float* __restrict__ out, int N)
{
    const int l = threadIdx.x;
    const bool has2 = (l < 16);
    const int j0 = l, j1 = 32 + l, j2 = 64 + l;

    const float wi00 = Wih0[j0 * 2 + 0], wi01 = Wih0[j0 * 2 + 1];
    const float wi10 = Wih0[j1 * 2 + 0], wi11 = Wih0[j1 * 2 + 1];
    const float wi20 = has2 ? Wih0[j2 * 2 + 0] : 0.0f;
    const float wi21 = has2 ? Wih0[j2 * 2 + 1] : 0.0f;
    const float b0 = bih0[j0] + bhh0[j0];
    const float b1 = bih0[j1] + bhh0[j1];
    const float b2 = has2 ? (bih0[j2] + bhh0[j2]) : 0.0f;

    float wh0[HD], wh1[HD], wh2[HD];
#pragma unroll
    for (int k = 0; k < HD; ++k) {
        wh0[k] = Whh0[j0 * HD + k];
        wh1[k] = Whh0[j1 * HD + k];
        wh2[k] = has2 ? Whh0[j2 * HD + k] : 0.0f;
    }

    float hr[HD];
#pragma unroll
    for (int k = 0; k < HD; ++k) hr[k] = 0.0f;
    float cst = 0.0f, hn = 0.0f;
    const int u = l;  // hidden unit owned by this lane (valid for l<20)

    for (int t = 0; t < N; ++t) {
        float g = grad[t];
        float lg = __logf(fabsf(g) + 1e-8f) * 0.1f;           // log(|g|+eps)/P, P=10
        lg = fminf(fmaxf(lg, -1.0f), 1.0f);
        float sg = fminf(fmaxf(g * 22026.465794806718f, -1.0f), 1.0f);  // g*e^10, clipped

        float a0 = fmaf(wi00, lg, fmaf(wi01, sg, b0));
        float a1 = fmaf(wi10, lg, fmaf(wi11, sg, b1));
        float a2 = fmaf(wi20, lg, fmaf(wi21, sg, b2));
#pragma unroll
        for (int k = 0; k < HD; ++k) {
            a0 = fmaf(wh0[k], hr[k], a0);
            a1 = fmaf(wh1[k], hr[k], a1);
            a2 = fmaf(wh2[k], hr[k], a2);
        }

        // gather raw gates i,f,g,o for hidden unit u (gate j -> lane j&31, slot j>>5)
        float iu = __shfl(a0, u, 32);                       // j = u        (<32)
        float f_lo = __shfl(a0, (20 + u) & 31, 32);         // j = 20+u, slot0 (u<12)
        float f_hi = __shfl(a1, (u - 12) & 31, 32);         // j = 20+u, slot1 (u>=12)
        float fu = (u < 12) ? f_lo : f_hi;
        float gu = __shfl(a1, (8 + u) & 31, 32);            // j = 40+u, slot1 always
        float o_lo = __shfl(a1, (28 + u) & 31, 32);         // j = 60+u, slot1 (u<4)
        float o_hi = __shfl(a2, (u - 4) & 31, 32);          // j = 60+u, slot2 (u>=4)
        float ou = (u < 4) ? o_lo : o_hi;

        float ia = sigm_f(iu), fa = sigm_f(fu), ga = tanh_f(gu), oa = sigm_f(ou);
        cst = fmaf(fa, cst, ia * ga);
        hn = oa * tanh_f(cst);

        if (l < HD) y0[t * HD + l] = hn;

#pragma unroll
        for (int k = 0; k < HD; ++k) hr[k] = __shfl(hn, k, 32);  // re-replicate h
    }

    if (l < HD) {
        out[N + l] = hn;            // h0
        out[N + 2 * HD + l] = cst;  // c0
    }
}

// ---------------------------------------------------------------------------
// Kernel 2: xproj1[N,80] = y0[N,20] @ Wih1.T[20,80] + (bih1+bhh1)
// via V_WMMA_F32_16X16X4_F32: M-tile = 16 timesteps, N=5x16 gates, K=5x4.
// One wave per 16-row strip; 25 wmma per block; bias preloaded into C.
// ---------------------------------------------------------------------------
__global__ void __launch_bounds__(32)
xproj1_wmma(const float* __restrict__ y0,
            const float* __restrict__ Wih1,
            const float* __restrict__ bih1, const float* __restrict__ bhh1,
            float* __restrict__ xproj, int N)
{
    const int t0 = blockIdx.x * 16;
    const int lane = threadIdx.x;
    const int half = lane >> 4;   // 0: lanes 0-15, 1: lanes 16-31
    const int m16 = lane & 15;

    // A (16x4 f32, 2 VGPRs): lane L<16 -> {A[L,kc*4+0], A[L,kc*4+1]},
    //                        lane L>=16 -> {A[L-16,kc*4+2], A[L-16,kc*4+3]}
    v2f a[5];
    const float* arow = y0 + (size_t)(t0 + m16) * HD;
#pragma unroll
    for (int kc = 0; kc < 5; ++kc) {
        int kk = kc * 4 + half * 2;
        a[kc].x = arow[kk];
        a[kc].y = arow[kk + 1];
    }

#pragma unroll
    for (int jt = 0; jt < 5; ++jt) {
        const int n = jt * 16 + m16;          // gate column this lane owns in B/C/D
        const float bias = bih1[n] + bhh1[n];
        v8f c;
#pragma unroll
        for (int r = 0; r < 8; ++r) c[r] = bias;

#pragma unroll
        for (int kc = 0; kc < 5; ++kc) {
            // B (4x16, 2 VGPRs): VGPR r = rows K=r (lanes 0-15) / K=2+r (lanes 16-31)
            int kb = kc * 4 + half * 2;
            v2f b;
            b.x = Wih1[n * HD + kb];
            b.y = Wih1[n * HD + kb + 1];
            c = __builtin_amdgcn_wmma_f32_16x16x4_f32(
                    false, a[kc], false, b, (short)0, c, false, false);
        }

        // D (16x16 f32): VGPR r -> row r + 8*half, col m16
#pragma unroll
        for (int r = 0; r < 8; ++r) {
            int m = half * 8 + r;
            xproj[(size_t)(t0 + m) * GATES + n] = c[r];
        }
    }
}

// ---------------------------------------------------------------------------
// Kernel 3: layer-1 LSTM scan + fused output linear, one wave32.
// ---------------------------------------------------------------------------
__global__ void __launch_bounds__(32)
lstm_scan1(const float* __restrict__ xproj,
           const float* __restrict__ Whh1,
           const float* __restrict__ Wlin, const float* __restrict__ blin,
           float* __restrict__ out, int N)
{
    const int l = threadIdx.x;
    const bool has2 = (l < 16);
    const int j0 = l, j1 = 32 + l, j2 = 64 + l;

    float wh0[HD], wh1v[HD], wh2[HD], wl[HD];
#pragma unroll
    for (int k = 0; k < HD; ++k) {
        wh0[k] = Whh1[j0 * HD + k];
        wh1v[k] = Whh1[j1 * HD + k];
        wh2[k] = has2 ? Whh1[j2 * HD + k] : 0.0f;
        wl[k] = Wlin[k];
    }
    const float bl = blin[0];

    float hr[HD];
#pragma unroll
    for (int k = 0; k < HD; ++k) hr[k] = 0.0f;
    float cst = 0.0f, hn = 0.0f;
    const int u = l;

    for (int t = 0; t < N; ++t) {
        const float* xp = xproj + (size_t)t * GATES;
        float a0 = xp[j0];
        float a1 = xp[j1];
        float a2 = has2 ? xp[j2] : 0.0f;
#pragma unroll
        for (int k = 0; k < HD; ++k) {
            a0 = fmaf(wh0[k], hr[k], a0);
            a1 = fmaf(wh1v[k], hr[k], a1);
            a2 = fmaf(wh2[k], hr[k], a2);
        }

        float iu = __shfl(a0, u, 32);
        float f_lo = __shfl(a0, (20 + u) & 31, 32);
        float f_hi = __shfl(a1, (u - 12) & 31, 32);
        float fu = (u < 12) ? f_lo : f_hi;
        float gu = __shfl(a1, (8 + u) & 31, 32);
        float o_lo = __shfl(a1, (28 + u) & 31, 32);
        float o_hi = __shfl(a2, (u - 4) & 31, 32);
        float ou = (u < 4) ? o_lo : o_hi;

        float ia = sigm_f(iu), fa = sigm_f(fu), ga = tanh_f(gu), oa = sigm_f(ou);
        cst = fmaf(fa, cst, ia * ga);
        hn = oa * tanh_f(cst);

#pragma unroll
        for (int k = 0; k < HD; ++k) hr[k] = __shfl(hn, k, 32);

        // fused output linear: every lane holds full h after broadcast
        float upd = bl;
#pragma unroll
        for (int k = 0; k < HD; ++k) upd = fmaf(wl[k], hr[k], upd);
        if (l == 0) out[t] = upd * 0.01f;
    }

    if (l < HD) {
        out[N + HD + l] = hn;       // h1
        out[N + 3 * HD + l] = cst;  // c1
    }
}

// ---------------------------------------------------------------------------
// Host launcher
// ---------------------------------------------------------------------------
extern "C" void kernel_launch(void* const* d_in, const int* in_sizes, int n_in,
                              void* d_out, int out_size, void* d_ws, size_t ws_size,
                              hipStream_t stream)
{
    const float* grad = (const float*)d_in[0];
    const float* Wih0 = (const float*)d_in[1];
    const float* Whh0 = (const float*)d_in[2];
    const float* bih0 = (const float*)d_in[3];
    const float* bhh0 = (const float*)d_in[4];
    const float* Wih1 = (const float*)d_in[5];
    const float* Whh1 = (const float*)d_in[6];
    const float* bih1 = (const float*)d_in[7];
    const float* bhh1 = (const float*)d_in[8];
    const float* Wlin = (const float*)d_in[9];
    const float* blin = (const float*)d_in[10];
    float* out = (float*)d_out;
    const int N = in_sizes[0];  // 65536

    // workspace: y0[N*HD] then xproj1[N*GATES]
    float* y0 = (float*)d_ws;
    float* xproj = y0 + (size_t)N * HD;

    hipLaunchKernelGGL(lstm_scan0, dim3(1), dim3(32), 0, stream,
                       grad, Wih0, Whh0, bih0, bhh0, y0, out, N);
    hipLaunchKernelGGL(xproj1_wmma, dim3(N / 16), dim3(32), 0, stream,
                       y0, Wih1, bih1, bhh1, xproj, N);
    hipLaunchKernelGGL(lstm_scan1, dim3(1), dim3(32), 0, stream,
                       xproj, Whh1, Wlin, blin, out, N);
}